// RandomSampling_52785148067994
// MI455X (gfx1250) — compile-verified
//
#include <hip/hip_runtime.h>

// ---------------- tuning constants ----------------
#define CHUNK     3072                 // elements per block in count/scatter (divides 30720)
#define CTHREADS  128                  // 4 wave32 waves
#define CITER     (CHUNK / CTHREADS)   // 24 strided iterations
#define NWAVE     (CTHREADS / 32)      // 4
#define NGRP      (CITER * NWAVE)      // 96 ballot groups per block

#define REDUCTION_INV 0.1f
#define EPSILON       1e-7f

typedef float v4f __attribute__((vector_size(4 * sizeof(float))));  // raw vec for NT ops

// ---- CDNA5 async global<->LDS path (ASYNCcnt-tracked DMA), guarded ----
#if __has_builtin(__builtin_amdgcn_global_load_async_to_lds_b128) && \
    __has_builtin(__builtin_amdgcn_global_store_async_from_lds_b128)
#define ASYNC_LDS 1
typedef int v4i __attribute__((vector_size(4 * sizeof(int))));
typedef __attribute__((address_space(1))) v4i gas_v4i;   // global int4
typedef __attribute__((address_space(3))) v4i las_v4i;   // LDS int4
#define AS1(p) ((gas_v4i*)(p))
#define AS3(p) ((las_v4i*)(p))
#else
#define ASYNC_LDS 0
#endif

__device__ __forceinline__ void wait_async0() {
#if __has_builtin(__builtin_amdgcn_s_wait_asynccnt)
  __builtin_amdgcn_s_wait_asynccnt(0);
#else
  asm volatile("s_wait_asynccnt 0x0" ::: "memory");
#endif
}

// wave32 ballot (gfx1250 is wave32-only)
__device__ __forceinline__ unsigned ballot32(bool p) {
#if __has_builtin(__builtin_amdgcn_ballot_w32)
  return __builtin_amdgcn_ballot_w32(p);
#else
  return (unsigned)__ballot((int)p);
#endif
}

// ---------------- K1: per-segment min of masked score ----------------
__global__ void seg_min_kernel(const float* __restrict__ score,
                               const int*   __restrict__ is_track,
                               const int*   __restrict__ row_splits,
                               float*       __restrict__ segmin) {
  __shared__ float red[256];
  const int b   = blockIdx.x;
  const int beg = row_splits[b];
  const int end = row_splits[b + 1];
  float m = 3.402823466e38f;
  for (int i = beg + (int)threadIdx.x; i < end; i += 256) {
    float s = is_track[i] ? 0.0f : score[i];
    m = fminf(m, s);
  }
  red[threadIdx.x] = m;
  __syncthreads();
  for (int off = 128; off > 0; off >>= 1) {
    if ((int)threadIdx.x < off)
      red[threadIdx.x] = fminf(red[threadIdx.x], red[threadIdx.x + off]);
    __syncthreads();
  }
  if (threadIdx.x == 0) segmin[b] = red[0];
}

// ---------------- K2: threshold = select(max(segmin)); zero segcnt ----------------
__global__ void threshold_kernel(const float* __restrict__ segmin, int B,
                                 float* __restrict__ thr,
                                 int*   __restrict__ segcnt) {
  __shared__ float red[256];
  float m = -3.402823466e38f;
  for (int i = threadIdx.x; i < B; i += 256) m = fmaxf(m, segmin[i]);
  red[threadIdx.x] = m;
  __syncthreads();
  for (int off = 128; off > 0; off >>= 1) {
    if ((int)threadIdx.x < off)
      red[threadIdx.x] = fmaxf(red[threadIdx.x], red[threadIdx.x + off]);
    __syncthreads();
  }
  if (threadIdx.x == 0) {
    float hm = red[0];
    *thr = (hm < REDUCTION_INV) ? REDUCTION_INV : (hm + EPSILON);
  }
  for (int i = threadIdx.x; i < B; i += 256) segcnt[i] = 0;
}

// ---------------- K3: per-block selected count + per-segment histogram ----------------
__global__ void count_kernel(const float* __restrict__ score,
                             const int*   __restrict__ is_track,
                             const int*   __restrict__ row_splits,
                             int B, int N,
                             const float* __restrict__ thr,
                             int* __restrict__ blkcnt,
                             int* __restrict__ segcnt) {
  extern __shared__ int smem[];          // rs[B+1] | hist[B]
  int* rs   = smem;
  int* hist = smem + (B + 1);
  __shared__ int red[CTHREADS];

  for (int i = threadIdx.x; i < B + 1; i += CTHREADS) rs[i] = row_splits[i];
  for (int i = threadIdx.x; i < B;     i += CTHREADS) hist[i] = 0;
  __syncthreads();

  const float t   = *thr;
  const int  base = (int)blockIdx.x * CHUNK;
  int mycnt = 0;
  int runseg = -1, runcnt = 0;           // run-length aggregate LDS atomics

  for (int k = 0; k < CITER; ++k) {
    int i = base + k * CTHREADS + (int)threadIdx.x;
    if (i < N) {
      float s = is_track[i] ? 0.0f : score[i];
      if (s <= t) {
        ++mycnt;
        int lo = 0, hi = B;              // find seg: rs[seg] <= i < rs[seg+1]
        while (hi - lo > 1) { int mid = (lo + hi) >> 1; if (rs[mid] <= i) lo = mid; else hi = mid; }
        if (lo == runseg) { ++runcnt; }
        else { if (runcnt) atomicAdd(&hist[runseg], runcnt); runseg = lo; runcnt = 1; }
      }
    }
  }
  if (runcnt) atomicAdd(&hist[runseg], runcnt);

  red[threadIdx.x] = mycnt;
  __syncthreads();
  for (int off = CTHREADS / 2; off > 0; off >>= 1) {
    if ((int)threadIdx.x < off) red[threadIdx.x] += red[threadIdx.x + off];
    __syncthreads();
  }
  if (threadIdx.x == 0) blkcnt[blockIdx.x] = red[0];
  for (int i = threadIdx.x; i < B; i += CTHREADS)
    if (hist[i]) atomicAdd(&segcnt[i], hist[i]);
}

// ---------------- K4: single-block scans; writes new_rs / old_size / n_sel ----------------
__global__ void scan_kernel(const int* __restrict__ blkcnt, int nblk,
                            int* __restrict__ blkoff, int* __restrict__ nselp,
                            const int* __restrict__ segcnt, int B,
                            float* __restrict__ out_rs,
                            float* __restrict__ out_oldsize,
                            float* __restrict__ out_nsel, int N) {
  __shared__ int lds[1024];
  const int tid = threadIdx.x;

  // scan 1: block counts -> exclusive block offsets, total -> n_sel
  int v = (tid < nblk) ? blkcnt[tid] : 0;
  lds[tid] = v;
  __syncthreads();
  for (int off = 1; off < 1024; off <<= 1) {
    int t = (tid >= off) ? lds[tid - off] : 0;
    __syncthreads();
    lds[tid] += t;
    __syncthreads();
  }
  int total = lds[1023];
  if (tid < nblk) blkoff[tid] = lds[tid] - v;
  if (tid == 0) { *nselp = total; *out_nsel = (float)total; *out_oldsize = (float)N; }
  __syncthreads();

  // scan 2: segment counts -> new_rs (B+1 values)
  int v2 = (tid < B) ? segcnt[tid] : 0;
  lds[tid] = v2;
  __syncthreads();
  for (int off = 1; off < 1024; off <<= 1) {
    int t = (tid >= off) ? lds[tid - off] : 0;
    __syncthreads();
    lds[tid] += t;
    __syncthreads();
  }
  if (tid < B) out_rs[tid + 1] = (float)lds[tid];  // inclusive = new_rs[tid+1]
  if (tid == 0) out_rs[0] = 0.0f;
}

// ---------------- K5: stable partition scatter ----------------
// Selected rows move global->LDS->global via the CDNA5 async DMA engine
// (ASYNCcnt); zero rows are async-stored from a single shared zeroed LDS row.
__global__ void scatter_kernel(const float* __restrict__ x,
                               const float* __restrict__ score,
                               const int*   __restrict__ is_track,
                               int N, int F4,
                               const float* __restrict__ thr,
                               const int*   __restrict__ blkoff,
                               const int*   __restrict__ nselp,
                               float* __restrict__ out_x,
                               float* __restrict__ out_idx) {
  __shared__ int ssel[NGRP], sact[NGRP];   // group counts -> exclusive offsets
  __shared__ int gsel[128],  gact[128];    // scan buffers (NGRP=96 <= 128)
#if ASYNC_LDS
  __shared__ __align__(16) char zrow[128];                     // shared zero row
  __shared__ __align__(16) char stage[2][CTHREADS * 128];      // double-buffered slots
#endif

  const int tid  = threadIdx.x;
  const int wave = tid >> 5;
  const int lane = tid & 31;
  const unsigned lt = (1u << lane) - 1u;
  const float t  = *thr;
  const int base = (int)blockIdx.x * CHUNK;
  const int F    = F4 * 4;

#if ASYNC_LDS
  if (tid < 32) ((float*)zrow)[tid] = 0.0f;
#endif

  unsigned mbits = 0, abits = 0;

  // Phase A: ballots + prefetch of selected x rows (hides gather latency)
  for (int k = 0; k < CITER; ++k) {
    int i = base + k * CTHREADS + tid;
    bool active = (i < N);
    bool m = false;
    if (active) {
      float s = is_track[i] ? 0.0f : score[i];
      m = (s <= t);
      if (m) {
        const char* xr = (const char*)(x + (size_t)i * F);
        __builtin_prefetch(xr, 0, 1);          // global_prefetch_b8
        __builtin_prefetch(xr + 64, 0, 1);
      }
    }
    unsigned bm = ballot32(m);
    unsigned ba = ballot32(active);
    if (lane == 0) {
      ssel[k * NWAVE + wave] = __popc(bm);
      sact[k * NWAVE + wave] = __popc(ba);
    }
    mbits |= (unsigned)m << k;
    abits |= (unsigned)active << k;
  }
  __syncthreads();

  // Phase B: exclusive scans over the 96 group counts
  int vs = (tid < NGRP) ? ssel[tid] : 0;
  int va = (tid < NGRP) ? sact[tid] : 0;
  gsel[tid] = vs;
  gact[tid] = va;
  __syncthreads();
  for (int off = 1; off < 128; off <<= 1) {
    int ts = 0, ta = 0;
    if (tid >= off) { ts = gsel[tid - off]; ta = gact[tid - off]; }
    __syncthreads();
    gsel[tid] += ts; gact[tid] += ta;
    __syncthreads();
  }
  if (tid < NGRP) { ssel[tid] = gsel[tid] - vs; sact[tid] = gact[tid] - va; }
  __syncthreads();

  // Phase C: scatter. Each active element writes exactly one output row.
  const int selbase   = blkoff[blockIdx.x];
  const int nsel      = *nselp;
  const int unselbase = nsel + base - selbase;   // unselected before this block

  for (int k = 0; k < CITER; ++k) {
    bool m      = (mbits >> k) & 1u;
    bool active = (abits >> k) & 1u;
    unsigned bm = ballot32(m);
    unsigned ba = ballot32(active);
    int g = k * NWAVE + wave;
    int i = base + k * CTHREADS + tid;
    int psel = selbase + ssel[g] + __popc(bm & lt);
    int puns = unselbase + (sact[g] - ssel[g]) + (__popc(ba & lt) - __popc(bm & lt));

#if ASYNC_LDS
    char* slot = &stage[k & 1][(unsigned)tid * 128u];
    if (m) {
      // async b128 loads: global row -> this lane's LDS slot (ASYNCcnt)
      const char* src = (const char*)(x + (size_t)i * F);
      for (int q = 0; q < F4; ++q)
        __builtin_amdgcn_global_load_async_to_lds_b128(AS1(src + q * 16),
                                                       AS3(slot + q * 16), 0, 0);
    }
    // Drains this iteration's loads AND iteration k-1's stores, so reusing the
    // k-parity slot at k+2 (issued after the wait at k+1) is hazard-free,
    // while stores(k-1) still overlap loads(k).
    wait_async0();
    if (m) {
      char* dst = (char*)(out_x + (size_t)psel * F);
      for (int q = 0; q < F4; ++q)
        __builtin_amdgcn_global_store_async_from_lds_b128(AS1(dst + q * 16),
                                                          AS3(slot + q * 16), 0, 0);
      out_idx[psel] = (float)i;
    } else if (active) {
      // zero fill: every unselected lane async-stores from the same zeroed row
      char* dst = (char*)(out_x + (size_t)puns * F);
      for (int q = 0; q < F4; ++q)
        __builtin_amdgcn_global_store_async_from_lds_b128(AS1(dst + q * 16),
                                                          AS3((char*)zrow + q * 16), 0, 0);
      out_idx[puns] = -1.0f;
    }
#else
    if (m) {
      const v4f* src = (const v4f*)(x + (size_t)i * F);
      v4f* dst = (v4f*)(out_x + (size_t)psel * F);
#pragma unroll 8
      for (int q = 0; q < F4; ++q) {
        v4f v = __builtin_nontemporal_load(src + q);
        __builtin_nontemporal_store(v, dst + q);
      }
      out_idx[psel] = (float)i;
    } else if (active) {
      v4f z = {0.f, 0.f, 0.f, 0.f};
      v4f* dst = (v4f*)(out_x + (size_t)puns * F);
#pragma unroll 8
      for (int q = 0; q < F4; ++q) __builtin_nontemporal_store(z, dst + q);
      out_idx[puns] = -1.0f;
    }
#endif
  }
  // outstanding async stores are drained by S_ENDPGM's implicit wait-idle
}

// ---------------- host launcher ----------------
extern "C" void kernel_launch(void* const* d_in, const int* in_sizes, int n_in,
                              void* d_out, int out_size, void* d_ws, size_t ws_size,
                              hipStream_t stream) {
  const float* x          = (const float*)d_in[0];
  const float* score      = (const float*)d_in[1];
  const int*   is_track   = (const int*)  d_in[2];
  const int*   row_splits = (const int*)  d_in[3];

  const int NF = in_sizes[0];
  const int N  = in_sizes[1];
  const int B  = in_sizes[3] - 1;
  const int F  = NF / N;          // 32
  const int F4 = F / 4;           // 8
  int nblk = (N + CHUNK - 1) / CHUNK;   // 640 for this workload (<= 1024 required)

  // workspace layout (ints/floats are 4B words)
  float* segmin = (float*)d_ws;                    // [0 .. 1024)
  float* thr    = segmin + 1024;                   // 1 float
  int*   nselp  = (int*)d_ws + 1025;               // 1 int
  int*   segcnt = (int*)d_ws + 2048;               // B ints
  int*   blkcnt = (int*)d_ws + 4096;               // nblk ints
  int*   blkoff = (int*)d_ws + 4096 + 1024;        // nblk ints

  // output layout: x_sel | new_rs | old_size | indices | n_sel
  float* out_x    = (float*)d_out;
  float* out_rs   = out_x + (size_t)N * F;
  float* out_old  = out_rs + (B + 1);
  float* out_idx  = out_old + 1;
  float* out_nsel = out_idx + N;

  seg_min_kernel<<<B, 256, 0, stream>>>(score, is_track, row_splits, segmin);
  threshold_kernel<<<1, 256, 0, stream>>>(segmin, B, thr, segcnt);
  count_kernel<<<nblk, CTHREADS, (size_t)(2 * B + 1) * sizeof(int), stream>>>(
      score, is_track, row_splits, B, N, thr, blkcnt, segcnt);
  scan_kernel<<<1, 1024, 0, stream>>>(blkcnt, nblk, blkoff, nselp,
                                      segcnt, B, out_rs, out_old, out_nsel, N);
  scatter_kernel<<<nblk, CTHREADS, 0, stream>>>(x, score, is_track, N, F4, thr,
                                                blkoff, nselp, out_x, out_idx);
}